// pyramid_step_model_85873576116776
// MI455X (gfx1250) — compile-verified
//
#include <hip/hip_runtime.h>
#include <hip/hip_bf16.h>
#include <math.h>

// ---------------------------------------------------------------------------
// Problem constants (from reference)
// ---------------------------------------------------------------------------
#define B_    4
#define S_    16384
#define E_    16
#define NX_   64
#define NY_   64
#define T_    (NX_ * NY_)          // 4096
#define NH_   8
#define MD_NH_ 32
#define FF_NH_ 64
#define N_    8192
#define NRES_ 90
#define NHI_  5
#define SIG_  0.5f

typedef float v2f __attribute__((ext_vector_type(2)));
typedef float v8f __attribute__((ext_vector_type(8)));

// ---------------------------------------------------------------------------
// Kernel A: top-8 nearest source points per grid point.
// One wave (32 lanes) per grid point; each lane keeps a sorted top-8 of
// 64-bit keys (d2 bits << 32 | source index) -> exact (d2, idx) ordering,
// ties to lowest index like jax.lax.top_k on -d2.
// ---------------------------------------------------------------------------
__global__ __launch_bounds__(256)
void knn_top8_kernel(const float* __restrict__ csrc,   // (2, S)
                     const float* __restrict__ cgrid,  // (2, T)
                     int* __restrict__ idx_out)        // (T, NH)
{
    const int lane = threadIdx.x & 31;
    const int wv   = threadIdx.x >> 5;
    const int t    = blockIdx.x * 8 + wv;   // 512 blocks * 8 waves = 4096

    const float gx = cgrid[t];
    const float gy = cgrid[T_ + t];

    unsigned long long best[8];
#pragma unroll
    for (int j = 0; j < 8; ++j) best[j] = ~0ull;

    for (int s = lane; s < S_; s += 32) {
        // CDNA5 speculative prefetch of upcoming source coords
        __builtin_prefetch(csrc + s + 512, 0, 0);
        const float dx = gx - csrc[s];
        const float dy = gy - csrc[S_ + s];
        const float d2 = dx * dx + dy * dy;
        const unsigned long long key =
            ((unsigned long long)__float_as_uint(d2) << 32) | (unsigned)s;
        if (key < best[7]) {
            best[7] = key;
#pragma unroll
            for (int j = 7; j > 0; --j) {   // single bubble pass (insert)
                unsigned long long lo = best[j - 1] < best[j] ? best[j - 1] : best[j];
                unsigned long long hi = best[j - 1] < best[j] ? best[j] : best[j - 1];
                best[j - 1] = lo;
                best[j]     = hi;
            }
        }
    }

    // Cross-lane merge: 8 rounds of wave-min over each lane's current head.
    int pos = 0;
#pragma unroll
    for (int k = 0; k < 8; ++k) {
        unsigned long long cand = (pos < 8) ? best[pos] : ~0ull;
        unsigned long long m = cand;
#pragma unroll
        for (int d = 16; d >= 1; d >>= 1) {
            unsigned long long o = __shfl_xor(m, d, 32);
            m = (o < m) ? o : m;
        }
        if (cand == m && pos < 8) ++pos;   // winner pops its head (keys unique)
        if (lane == 0) idx_out[t * NH_ + k] = (int)(m & 0xffffffffu);
    }
}

// ---------------------------------------------------------------------------
// Kernel B: PE MLP + LayerNorm + logits + softmax + neighbor-weighted gather.
// One wave handles 2 grid points = 16 rows of (t, n).
//  - pe_w2 (8KB) and k_w (8KB) are staged into LDS once per workgroup using
//    gfx1250 GLOBAL_LOAD_ASYNC_TO_LDS_B128 (ASYNCcnt), overlapped with the
//    per-lane layer-1 gelu work, fenced by s_wait_asynccnt + barrier.
//  - Layer 1 (K=2) is computed per-lane directly in WMMA A-operand layout.
//  - Layer 2 is 16 K-steps x 2 N-tiles of v_wmma_f32_16x16x4_f32, with the
//    B operands read from LDS.
// ---------------------------------------------------------------------------
__device__ __forceinline__ float gelu_tanh(float x) {
    const float k0 = 0.7978845608028654f;  // sqrt(2/pi)
    return 0.5f * x * (1.0f + tanhf(k0 * (x + 0.044715f * x * x * x)));
}

__device__ __forceinline__ unsigned lds_addr_of(const void* p) {
    // LDS flat aperture maps addr[31:0] directly to the LDS byte address.
    return (unsigned)(uintptr_t)p;
}

__device__ __forceinline__ void async_copy_b128(unsigned lds_byte_addr,
                                                unsigned glob_byte_off,
                                                const void* glob_base) {
    asm volatile("global_load_async_to_lds_b128 %0, %1, %2"
                 :: "v"(lds_byte_addr), "v"(glob_byte_off), "s"(glob_base)
                 : "memory");
}

__global__ __launch_bounds__(128)
void grid_assemble_kernel(const float* __restrict__ x,      // (B, S, E)
                          const float* __restrict__ csrc,   // (2, S)
                          const float* __restrict__ cgrid,  // (2, T)
                          const float* __restrict__ pe_w1,  // (2, 64)
                          const float* __restrict__ pe_b1,  // (64)
                          const float* __restrict__ pe_w2,  // (64, 32)
                          const float* __restrict__ pe_b2,  // (32)
                          const float* __restrict__ ln_g,   // (32)
                          const float* __restrict__ ln_b,   // (32)
                          const float* __restrict__ k_w,    // (256, 8)
                          const float* __restrict__ k_b,    // (8)
                          const int* __restrict__ idx,      // (T, NH)
                          float* __restrict__ gbuf)         // (B, T, E)
{
    __shared__ float s_w2[FF_NH_ * MD_NH_];   // 64x32 weights (8KB)
    __shared__ float s_kw[256 * NH_];         // 256x8 weights (8KB)
    __shared__ float s_kf[4][16 * MD_NH_];    // per-wave 16x32 kf tile
    __shared__ float s_wt[4][16];             // per-wave softmax weights

    const int tid  = threadIdx.x;
    const int lane = threadIdx.x & 31;
    const int wv   = threadIdx.x >> 5;
    const int lo   = lane & 15;   // column-within-tile / row id
    const int hi   = lane >> 4;   // lane half -> K sub-pair selector
    const int t0   = (blockIdx.x * 4 + wv) * 2;   // 2 grid points per wave

    // ---- async stage of shared weights into LDS (overlaps layer-1 VALU) ----
    {
        const unsigned b_w2 = lds_addr_of(&s_w2[0]);
        const unsigned b_kw = lds_addr_of(&s_kw[0]);
#pragma unroll
        for (int p = 0; p < 4; ++p) {            // 128 thr * 16B * 4 = 8KB each
            const unsigned off = (unsigned)(tid + p * 128) * 16u;
            async_copy_b128(b_w2 + off, off, pe_w2);
            async_copy_b128(b_kw + off, off, k_w);
        }
    }

    // ---- row -> (t, n) mapping for this lane's A-operand row ----
    const int r  = lo;                 // A-matrix row 0..15
    const int t  = t0 + (r >> 3);
    const int n  = r & 7;
    const int si = idx[t * NH_ + n];
    const float relx = csrc[si]       - cgrid[t];
    const float rely = csrc[S_ + si]  - cgrid[T_ + t];

    // ---- Layer 1 (K=2) directly into WMMA A layout: h[r][4k + 2*hi + u] ----
    float ha[16][2];
#pragma unroll
    for (int k = 0; k < 16; ++k) {
#pragma unroll
        for (int u = 0; u < 2; ++u) {
            const int c = 4 * k + 2 * hi + u;
            const float z = relx * pe_w1[c] + rely * pe_w1[FF_NH_ + c] + pe_b1[c];
            ha[k][u] = gelu_tanh(z);
        }
    }

    // ---- fence async weight staging before any LDS reads ----
    asm volatile("s_wait_asynccnt 0x0" ::: "memory");
    __syncthreads();

    // ---- Layer 2: pe(16x32) = h(16x64) @ W2(64x32) + b2, via WMMA ----
    v8f acc0, acc1;
    const float bias0 = pe_b2[lo];
    const float bias1 = pe_b2[16 + lo];
#pragma unroll
    for (int p = 0; p < 8; ++p) { acc0[p] = bias0; acc1[p] = bias1; }

#pragma unroll
    for (int k = 0; k < 16; ++k) {
        v2f a; a[0] = ha[k][0]; a[1] = ha[k][1];
        const int krow = 4 * k + 2 * hi;
        v2f b0; b0[0] = s_w2[krow * MD_NH_ + lo];
                b0[1] = s_w2[(krow + 1) * MD_NH_ + lo];
        v2f b1; b1[0] = s_w2[krow * MD_NH_ + 16 + lo];
                b1[1] = s_w2[(krow + 1) * MD_NH_ + 16 + lo];
        acc0 = __builtin_amdgcn_wmma_f32_16x16x4_f32(false, a, false, b0,
                                                     (short)0, acc0, false, false);
        acc1 = __builtin_amdgcn_wmma_f32_16x16x4_f32(false, a, false, b1,
                                                     (short)0, acc1, false, false);
    }

    // ---- LayerNorm over 32 cols of each row (rows p / p+8 in lane halves) ----
#pragma unroll
    for (int p = 0; p < 8; ++p) {
        float v0 = acc0[p], v1 = acc1[p];
        float s  = v0 + v1;
        float sq = v0 * v0 + v1 * v1;
#pragma unroll
        for (int d = 8; d >= 1; d >>= 1) {     // reduce within 16-lane half
            s  += __shfl_xor(s,  d, 16);
            sq += __shfl_xor(sq, d, 16);
        }
        const float mean = s * (1.0f / 32.0f);
        const float var  = sq * (1.0f / 32.0f) - mean * mean;
        const float rstd = rsqrtf(var + 1e-5f);
        const int   row  = p + 8 * hi;
        const float kf0  = (v0 - mean) * rstd * ln_g[lo]      + ln_b[lo];
        const float kf1  = (v1 - mean) * rstd * ln_g[16 + lo] + ln_b[16 + lo];
        s_kf[wv][row * MD_NH_ + lo]      = kf0;
        s_kf[wv][row * MD_NH_ + 16 + lo] = kf1;
    }
    __syncthreads();

    // ---- logits (per grid point: 256-dot with k_w) + softmax over NH=8 ----
    {
        const int g = (lane & 15) >> 3;   // grid point within tile (dup on hi half)
        const int o = lane & 7;           // output head
        const float* kfv = &s_kf[wv][g * 256];
        float lg = k_b[o];
#pragma unroll 8
        for (int q = 0; q < 256; ++q) lg += kfv[q] * s_kw[q * NH_ + o];

        float mx = lg;
#pragma unroll
        for (int d = 4; d >= 1; d >>= 1) mx = fmaxf(mx, __shfl_xor(mx, d, 8));
        float e = __expf(lg - mx);
        float sum = e;
#pragma unroll
        for (int d = 4; d >= 1; d >>= 1) sum += __shfl_xor(sum, d, 8);
        const float w = e / sum;
        if (lane < 16) s_wt[wv][lane] = w;   // [g*8 + o]
    }
    __syncthreads();

    // ---- xg[b][t][e] = sum_n w[t][n] * x[b][idx[t][n]][e]  (grid = B,T,E) ----
    {
        const int ech = lo;        // channel 0..15
        const int gl  = hi;        // grid point 0/1
        const int tt  = t0 + gl;
#pragma unroll
        for (int b = 0; b < B_; ++b) {
            float acc = 0.0f;
#pragma unroll
            for (int nn = 0; nn < NH_; ++nn) {
                const int ii = idx[tt * NH_ + nn];
                acc += s_wt[wv][gl * 8 + nn] * x[((size_t)b * S_ + ii) * E_ + ech];
            }
            gbuf[((size_t)b * T_ + tt) * E_ + ech] = acc;
        }
    }
}

// ---------------------------------------------------------------------------
// Kernel C: Gaussian-window separable interpolation onto N target points.
// One thread per (b, q); grid buffer is (B, NX, NY, E) so each of the 25 taps
// is a contiguous 64B read.
// ---------------------------------------------------------------------------
__global__ __launch_bounds__(256)
void target_interp_kernel(const float* __restrict__ ctgt,  // (B, N, 2)
                          const float* __restrict__ gbuf,  // (B, NX, NY, E)
                          float* __restrict__ out)         // (B, E, N)
{
    const int gid = blockIdx.x * blockDim.x + threadIdx.x;   // 0..B*N-1
    const int b = gid / N_;
    const int q = gid - b * N_;

    const float c0 = ctgt[((size_t)b * N_ + q) * 2 + 0];
    const float c1 = ctgt[((size_t)b * N_ + q) * 2 + 1];
    const float posx = c1 * (float)(NX_ - 1);
    const float posy = c0 * (float)(NY_ - 1);
    const float rpx = rintf(posx);
    const float rpy = rintf(posy);

    const float INVN = 0.7978845608028654f;  // 1/(SIG*sqrt(2*pi)), SIG=0.5
    float wx[NHI_], wy[NHI_];
#pragma unroll
    for (int a = 0; a < NHI_; ++a) { wx[a] = 0.0f; wy[a] = 0.0f; }

    for (int i = 0; i < NRES_; ++i) {
        const float off = 2.5f - 5.0f * (float)i / 89.0f;   // linspace(2.5,-2.5,90)
        const float pxv = fminf(fmaxf(rpx - off, 0.0f), (float)NX_);
        const float pyv = fminf(fmaxf(rpy - off, 0.0f), (float)NX_);
        const float tx = (pxv - posx) * 2.0f;               // /SIG
        const float ty = (pyv - posy) * 2.0f;
        const int g = i / (NRES_ / NHI_);                    // group of 18
        wx[g] += __expf(-0.5f * tx * tx) * INVN;
        wy[g] += __expf(-0.5f * ty * ty) * INVN;
    }

    int ixv[NHI_], iyv[NHI_];
    float sx = 0.0f, sy = 0.0f;
#pragma unroll
    for (int a = 0; a < NHI_; ++a) {
        const float offi = (float)(a - 2);                   // [-2..2]
        int ix = (int)rintf(rpx - offi);
        int iy = (int)rintf(rpy - offi);
        ixv[a] = min(max(ix, 0), NX_ - 1);
        iyv[a] = min(max(iy, 0), NX_ - 1);
        sx += wx[a]; sy += wy[a];
    }
    const float inv = 1.0f / (sx * sy);

    float acc[E_];
#pragma unroll
    for (int e = 0; e < E_; ++e) acc[e] = 0.0f;

#pragma unroll
    for (int a = 0; a < NHI_; ++a) {
#pragma unroll
        for (int c = 0; c < NHI_; ++c) {
            const float wt = wx[a] * wy[c];
            const float4* g4 = (const float4*)(gbuf +
                (((size_t)b * T_ + ixv[a] * NY_ + iyv[c]) * E_));
#pragma unroll
            for (int v = 0; v < 4; ++v) {
                const float4 gv = g4[v];
                acc[v * 4 + 0] += wt * gv.x;
                acc[v * 4 + 1] += wt * gv.y;
                acc[v * 4 + 2] += wt * gv.z;
                acc[v * 4 + 3] += wt * gv.w;
            }
        }
    }
#pragma unroll
    for (int e = 0; e < E_; ++e)
        out[((size_t)b * E_ + e) * N_ + q] = acc[e] * inv;
}

// ---------------------------------------------------------------------------
// Launch
// ---------------------------------------------------------------------------
extern "C" void kernel_launch(void* const* d_in, const int* in_sizes, int n_in,
                              void* d_out, int out_size, void* d_ws, size_t ws_size,
                              hipStream_t stream) {
    const float* x     = (const float*)d_in[0];
    const float* csrc  = (const float*)d_in[1];
    const float* cgrid = (const float*)d_in[2];
    const float* ctgt  = (const float*)d_in[3];
    const float* pe_w1 = (const float*)d_in[4];
    const float* pe_b1 = (const float*)d_in[5];
    const float* pe_w2 = (const float*)d_in[6];
    const float* pe_b2 = (const float*)d_in[7];
    const float* ln_g  = (const float*)d_in[8];
    const float* ln_b  = (const float*)d_in[9];
    const float* k_w   = (const float*)d_in[10];
    const float* k_b   = (const float*)d_in[11];
    float* out = (float*)d_out;

    // Workspace layout: [idx: T*NH int32][gbuf: B*T*E float]
    int*   idx  = (int*)d_ws;
    float* gbuf = (float*)((char*)d_ws + (size_t)T_ * NH_ * sizeof(int));

    // A: kNN top-8   (4096 grid points, 1 wave each, 8 waves/block)
    knn_top8_kernel<<<T_ / 8, 256, 0, stream>>>(csrc, cgrid, idx);

    // B: MLP + LN + softmax + gather  (2 grid points/wave, 4 waves/block)
    grid_assemble_kernel<<<T_ / 8, 128, 0, stream>>>(
        x, csrc, cgrid, pe_w1, pe_b1, pe_w2, pe_b2, ln_g, ln_b, k_w, k_b,
        idx, gbuf);

    // C: target interpolation (one thread per (b, q))
    target_interp_kernel<<<(B_ * N_) / 256, 256, 0, stream>>>(ctgt, gbuf, out);
}